// GATEncoder_3066606649990
// MI455X (gfx1250) — compile-verified
//
#include <hip/hip_runtime.h>
#include <math.h>

typedef __attribute__((ext_vector_type(2))) float v2f;
typedef __attribute__((ext_vector_type(8))) float v8f;

#define NNODES 50000
#define NEDGES 800000
#define ETOT   (NEDGES + NNODES)   // 850000 incl. self loops
#define DFEAT  256                 // feature width both layers

// ---------------------------------------------------------------- helpers

__device__ __forceinline__ void atomicMaxF(float* addr, float v) {
    // monotone int-bits ordering trick (works for mixed signs, init = -inf)
    if (v >= 0.0f)
        atomicMax((int*)addr, __float_as_int(v));
    else
        atomicMin((unsigned int*)addr, (unsigned int)__float_as_int(v));
}

__device__ __forceinline__ void edge_endpoints(const long long* __restrict__ ei,
                                               int e, int& src, int& dst) {
    if (e < NEDGES) {
        src = (int)ei[e];
        dst = (int)ei[NEDGES + e];
    } else {               // appended self loops
        src = e - NEDGES;
        dst = e - NEDGES;
    }
}

// ---------------------------------------------------------------- W repack
// Wp[q][n][r] = W[2q+r][n]  (q = k/2, r = k&1): makes each WMMA B-fragment a
// single aligned 8-byte load (global_load_b64) instead of two strided b32s.
__global__ void repack_B_pairs(const float* __restrict__ W, float* __restrict__ Wp,
                               int K, int N) {
    int i = blockIdx.x * blockDim.x + threadIdx.x;
    if (i >= K * N) return;
    int q = i >> 9;              // i / (N*2)
    int rem = i & 511;
    int n = rem >> 1;
    int r = rem & 1;
    Wp[i] = W[(size_t)(2 * q + r) * N + n];
}

// ---------------------------------------------------------------- GEMM (fp32 WMMA)
// C[M,N] = A[M,K] @ B[K,N] with B pre-packed pair-interleaved.
// Exact fp32 via V_WMMA_F32_16X16X4_F32; one wave owns a 16(M) x 64(N) tile.
__global__ void gemm_f32_wmma(const float* __restrict__ A,
                              const float* __restrict__ Bp,   // packed [K/2][N][2]
                              float* __restrict__ C,
                              int M, int N, int K) {
    int wave = (int)((blockIdx.x * blockDim.x + threadIdx.x) >> 5);
    int lane = threadIdx.x & 31;
    int ntiles = N >> 6;
    int totalWaves = (M >> 4) * ntiles;
    if (wave >= totalWaves) return;           // wave-uniform: EXEC all-ones inside

    int mt = wave / ntiles;
    int nt = wave % ntiles;
    int m0 = mt << 4;
    int n0 = nt << 6;
    int half = lane >> 4;                     // A frag: 0 -> K={0,1}, 1 -> K={2,3}
    int l15  = lane & 15;

    v8f acc0 = {}, acc1 = {}, acc2 = {}, acc3 = {};
    const float* Arow = A + (size_t)(m0 + l15) * K + 2 * half;
    // B frag base: pair index (k0/2 + half), column (n0 + l15); 2 floats contiguous
    const float* Bbase = Bp + (size_t)half * (2 * N) + (size_t)(n0 + l15) * 2;

    for (int k0 = 0; k0 < K; k0 += 4) {
        v2f a;
        a.x = Arow[k0];
        a.y = Arow[k0 + 1];
        const float* Bq = Bbase + (size_t)(k0 >> 1) * (2 * N);
        v2f b0v = *(const v2f*)(Bq);          // fragment for N-subtile 0
        v2f b1v = *(const v2f*)(Bq + 32);     // +16 columns * 2
        v2f b2v = *(const v2f*)(Bq + 64);
        v2f b3v = *(const v2f*)(Bq + 96);
        acc0 = __builtin_amdgcn_wmma_f32_16x16x4_f32(false, a, false, b0v, (short)0, acc0, false, false);
        acc1 = __builtin_amdgcn_wmma_f32_16x16x4_f32(false, a, false, b1v, (short)0, acc1, false, false);
        acc2 = __builtin_amdgcn_wmma_f32_16x16x4_f32(false, a, false, b2v, (short)0, acc2, false, false);
        acc3 = __builtin_amdgcn_wmma_f32_16x16x4_f32(false, a, false, b3v, (short)0, acc3, false, false);
    }

    // C/D layout: VGPR r -> row m0+r (lanes 0-15) / m0+r+8 (lanes 16-31)
    #pragma unroll
    for (int r = 0; r < 8; ++r) {
        float* Crow = C + (size_t)(m0 + r + 8 * half) * N + n0 + l15;
        Crow[0]  = acc0[r];
        Crow[16] = acc1[r];
        Crow[32] = acc2[r];
        Crow[48] = acc3[r];
    }
}

// ---------------------------------------------------------------- attention pieces

// es[n,h] = sum_c H[n,h*C+c]*a_src[h,c]  (and ed likewise), float4 loads
__global__ void attn_logits(const float* __restrict__ Hf,
                            const float* __restrict__ a_src,
                            const float* __restrict__ a_dst,
                            float* __restrict__ es, float* __restrict__ ed,
                            int n, int Hh, int C) {
    int i = blockIdx.x * blockDim.x + threadIdx.x;
    if (i >= n * Hh) return;
    int node = i / Hh, h = i % Hh;
    const float4* row = (const float4*)(Hf + (size_t)node * (Hh * C) + h * C);
    const float4* as  = (const float4*)(a_src + h * C);
    const float4* ad  = (const float4*)(a_dst + h * C);
    float s = 0.f, d = 0.f;
    int C4 = C >> 2;
    for (int c = 0; c < C4; ++c) {
        float4 v = row[c], a = as[c], b = ad[c];
        s += v.x * a.x + v.y * a.y + v.z * a.z + v.w * a.w;
        d += v.x * b.x + v.y * b.y + v.z * b.z + v.w * b.w;
    }
    es[i] = s;
    ed[i] = d;
}

__global__ void init_node(float* __restrict__ m, float* __restrict__ z, int count) {
    int i = blockIdx.x * blockDim.x + threadIdx.x;
    if (i < count) { m[i] = __int_as_float(0xFF800000); z[i] = 0.f; }  // -inf, 0
}

__global__ void init_out_bias(float* __restrict__ out, const float* __restrict__ bias,
                              int n, int D) {
    int i = blockIdx.x * blockDim.x + threadIdx.x;
    if (i < n * D) out[i] = bias[i % D];   // softmax weights sum to 1 -> bias folds in
}

__global__ void zero_f32(float* __restrict__ p, int count) {
    int i = blockIdx.x * blockDim.x + threadIdx.x;
    if (i < count) p[i] = 0.f;
}

// pass 1: e = leaky_relu(es[src]+ed[dst]); store; segment-max into m[dst]
__global__ void edge_logit_max(const long long* __restrict__ ei,
                               const float* __restrict__ es, const float* __restrict__ ed,
                               float* __restrict__ logit, float* __restrict__ m, int Hh) {
    int i = blockIdx.x * blockDim.x + threadIdx.x;
    if (i >= ETOT * Hh) return;
    int e = i / Hh, h = i % Hh;
    int src, dst; edge_endpoints(ei, e, src, dst);
    float v = es[src * Hh + h] + ed[dst * Hh + h];
    v = (v > 0.f) ? v : 0.2f * v;            // leaky_relu slope 0.2
    logit[i] = v;
    atomicMaxF(&m[dst * Hh + h], v);
}

// pass 2: p = exp(e - m[dst]); store; segment-sum into z[dst]
__global__ void edge_exp_sum(const long long* __restrict__ ei,
                             float* __restrict__ logit, const float* __restrict__ m,
                             float* __restrict__ z, int Hh) {
    int i = blockIdx.x * blockDim.x + threadIdx.x;
    if (i >= ETOT * Hh) return;
    int e = i / Hh, h = i % Hh;
    int src, dst; edge_endpoints(ei, e, src, dst);
    (void)src;
    float p = expf(logit[i] - m[dst * Hh + h]);
    logit[i] = p;
    atomicAdd(&z[dst * Hh + h], p);
}

// pass 3: alpha = p / z[dst]  (in place)
__global__ void edge_alpha(const long long* __restrict__ ei,
                           float* __restrict__ logit, const float* __restrict__ z, int Hh) {
    int i = blockIdx.x * blockDim.x + threadIdx.x;
    if (i >= ETOT * Hh) return;
    int e = i / Hh, h = i % Hh;
    int src, dst; edge_endpoints(ei, e, src, dst);
    (void)src;
    logit[i] = logit[i] / z[dst * Hh + h];
}

// pass 4: out[dst, j..j+3] += H[src, j..j+3] * alpha[e, j/C]
// float4 gather (global_load_b128) + 4 atomic f32 adds; L2-resident scatter.
__global__ void edge_aggregate4(const long long* __restrict__ ei,
                                const float* __restrict__ Hf,
                                const float* __restrict__ alpha,
                                float* __restrict__ out, int Hh, int C) {
    long long i = (long long)blockIdx.x * blockDim.x + threadIdx.x;
    int D  = Hh * C;
    int Dq = D >> 2;
    if (i >= (long long)ETOT * Dq) return;
    int e  = (int)(i / Dq);
    int j  = ((int)(i % Dq)) << 2;      // 4 consecutive features, same head (C%4==0)
    int h  = j / C;
    int src, dst; edge_endpoints(ei, e, src, dst);
    float a = alpha[e * Hh + h];
    float4 hv = *(const float4*)(Hf + (size_t)src * D + j);
    float* o = out + (size_t)dst * D + j;
    atomicAdd(o + 0, hv.x * a);
    atomicAdd(o + 1, hv.y * a);
    atomicAdd(o + 2, hv.z * a);
    atomicAdd(o + 3, hv.w * a);
}

// ---------------------------------------------------------------- batchnorm + elu

__global__ void bn_stats(const float* __restrict__ x,
                         float* __restrict__ sum, float* __restrict__ sumsq, int n) {
    int col = threadIdx.x;                    // 256 threads = 256 columns
    int r0 = blockIdx.x * 128;
    int rend = r0 + 128; if (rend > n) rend = n;
    float s = 0.f, ss = 0.f;
    for (int r = r0; r < rend; ++r) {
        float v = x[(size_t)r * DFEAT + col];
        s += v; ss += v * v;
    }
    atomicAdd(&sum[col], s);
    atomicAdd(&sumsq[col], ss);
}

__global__ void bn_elu(float* __restrict__ x,
                       const float* __restrict__ sum, const float* __restrict__ sumsq,
                       const float* __restrict__ gamma, const float* __restrict__ beta,
                       int n) {
    int i = blockIdx.x * blockDim.x + threadIdx.x;
    if (i >= n * DFEAT) return;
    int col = i & (DFEAT - 1);
    float invn = 1.0f / (float)n;
    float mu  = sum[col] * invn;
    float var = sumsq[col] * invn - mu * mu;   // biased variance
    float y = gamma[col] * (x[i] - mu) * rsqrtf(var + 1e-5f) + beta[col];
    x[i] = (y > 0.f) ? y : (expf(y) - 1.f);    // ELU (alpha=1)
}

// ---------------------------------------------------------------- final L2 normalize

__global__ void l2norm_rows(float* __restrict__ x, int n) {
    __shared__ float red[DFEAT];
    int row = blockIdx.x;
    if (row >= n) return;
    float v = x[(size_t)row * DFEAT + threadIdx.x];
    red[threadIdx.x] = v * v;
    __syncthreads();
    for (int s = DFEAT / 2; s > 0; s >>= 1) {
        if ((int)threadIdx.x < s) red[threadIdx.x] += red[threadIdx.x + s];
        __syncthreads();
    }
    float norm = fmaxf(sqrtf(red[0]), 1e-12f);
    x[(size_t)row * DFEAT + threadIdx.x] = v / norm;
}

// ---------------------------------------------------------------- launch

extern "C" void kernel_launch(void* const* d_in, const int* in_sizes, int n_in,
                              void* d_out, int out_size, void* d_ws, size_t ws_size,
                              hipStream_t stream) {
    (void)in_sizes; (void)n_in; (void)out_size; (void)ws_size;

    const float*     x      = (const float*)d_in[0];
    const long long* ei     = (const long long*)d_in[1];   // int64 [2, E]
    const float*     W0     = (const float*)d_in[2];
    const float*     a_src0 = (const float*)d_in[3];
    const float*     a_dst0 = (const float*)d_in[4];
    const float*     b0     = (const float*)d_in[5];
    const float*     gamma0 = (const float*)d_in[6];
    const float*     beta0  = (const float*)d_in[7];
    const float*     W1     = (const float*)d_in[8];
    const float*     a_src1 = (const float*)d_in[9];
    const float*     a_dst1 = (const float*)d_in[10];
    const float*     b1     = (const float*)d_in[11];
    float*           out    = (float*)d_out;

    // workspace carve-up (256B aligned)
    char* ws = (char*)d_ws;
    size_t off = 0;
    auto carve = [&](size_t bytes) { char* p = ws + off; off += (bytes + 255) & ~(size_t)255; return p; };
    float* Hbuf   = (float*)carve((size_t)NNODES * DFEAT * 4); // H0, reused as H1
    float* AGG    = (float*)carve((size_t)NNODES * DFEAT * 4); // layer0 agg -> GEMM1 input
    float* Wpack  = (float*)carve((size_t)DFEAT * DFEAT * 4);  // pair-interleaved W (per layer)
    float* es     = (float*)carve((size_t)NNODES * 4 * 4);
    float* ed     = (float*)carve((size_t)NNODES * 4 * 4);
    float* mbuf   = (float*)carve((size_t)NNODES * 4 * 4);
    float* zbuf   = (float*)carve((size_t)NNODES * 4 * 4);
    float* pbuf   = (float*)carve((size_t)ETOT * 4 * 4);       // logits -> p -> alpha
    float* colsum = (float*)carve(DFEAT * 4);
    float* colsq  = (float*)carve(DFEAT * 4);

    const int BT = 256;
    auto blocks = [](long long t, int b) { return (unsigned)((t + b - 1) / b); };

    int gemmWaves = (NNODES / 16) * (DFEAT / 64);              // 12500
    unsigned gemmBlocks = blocks((long long)gemmWaves * 32, BT);
    unsigned repackBlocks = blocks((long long)DFEAT * DFEAT, BT);
    unsigned agg4Blocks = blocks((long long)ETOT * (DFEAT / 4), BT);

    // ================= layer 0 (H=4, C=64) =================
    repack_B_pairs<<<repackBlocks, BT, 0, stream>>>(W0, Wpack, DFEAT, DFEAT);
    gemm_f32_wmma<<<gemmBlocks, BT, 0, stream>>>(x, Wpack, Hbuf, NNODES, DFEAT, DFEAT);
    attn_logits<<<blocks((long long)NNODES * 4, BT), BT, 0, stream>>>(Hbuf, a_src0, a_dst0, es, ed, NNODES, 4, 64);
    init_node<<<blocks((long long)NNODES * 4, BT), BT, 0, stream>>>(mbuf, zbuf, NNODES * 4);
    init_out_bias<<<blocks((long long)NNODES * DFEAT, BT), BT, 0, stream>>>(AGG, b0, NNODES, DFEAT);
    edge_logit_max<<<blocks((long long)ETOT * 4, BT), BT, 0, stream>>>(ei, es, ed, pbuf, mbuf, 4);
    edge_exp_sum<<<blocks((long long)ETOT * 4, BT), BT, 0, stream>>>(ei, pbuf, mbuf, zbuf, 4);
    edge_alpha<<<blocks((long long)ETOT * 4, BT), BT, 0, stream>>>(ei, pbuf, zbuf, 4);
    edge_aggregate4<<<agg4Blocks, BT, 0, stream>>>(ei, Hbuf, pbuf, AGG, 4, 64);
    zero_f32<<<blocks(DFEAT, BT), BT, 0, stream>>>(colsum, DFEAT);
    zero_f32<<<blocks(DFEAT, BT), BT, 0, stream>>>(colsq, DFEAT);
    bn_stats<<<blocks(NNODES, 128), DFEAT, 0, stream>>>(AGG, colsum, colsq, NNODES);
    bn_elu<<<blocks((long long)NNODES * DFEAT, BT), BT, 0, stream>>>(AGG, colsum, colsq, gamma0, beta0, NNODES);

    // ================= layer 1 (H=1, C=256) =================
    repack_B_pairs<<<repackBlocks, BT, 0, stream>>>(W1, Wpack, DFEAT, DFEAT);
    gemm_f32_wmma<<<gemmBlocks, BT, 0, stream>>>(AGG, Wpack, Hbuf, NNODES, DFEAT, DFEAT);
    attn_logits<<<blocks(NNODES, BT), BT, 0, stream>>>(Hbuf, a_src1, a_dst1, es, ed, NNODES, 1, 256);
    init_node<<<blocks(NNODES, BT), BT, 0, stream>>>(mbuf, zbuf, NNODES);
    init_out_bias<<<blocks((long long)NNODES * DFEAT, BT), BT, 0, stream>>>(out, b1, NNODES, DFEAT);
    edge_logit_max<<<blocks(ETOT, BT), BT, 0, stream>>>(ei, es, ed, pbuf, mbuf, 1);
    edge_exp_sum<<<blocks(ETOT, BT), BT, 0, stream>>>(ei, pbuf, mbuf, zbuf, 1);
    edge_alpha<<<blocks(ETOT, BT), BT, 0, stream>>>(ei, pbuf, zbuf, 1);
    edge_aggregate4<<<agg4Blocks, BT, 0, stream>>>(ei, Hbuf, pbuf, out, 1, 256);

    // ================= final row-wise L2 normalize =================
    l2norm_rows<<<NNODES, DFEAT, 0, stream>>>(out, NNODES);
}